// LinearKernelAttention_82592221102306
// MI455X (gfx1250) — compile-verified
//
#include <hip/hip_runtime.h>
#include <hip/hip_bf16.h>
#include <math.h>

typedef __attribute__((ext_vector_type(2))) float v2f;
typedef __attribute__((ext_vector_type(4))) float v4f;
typedef __attribute__((ext_vector_type(8))) float v8f;

#define SEQ 2048
#define DIM 64
#define LDS_STRIDE 2052   // 2048 + 4 pad: multiple of 4 (16B align), rows 8 apart -> 32 banks apart

// One workgroup (4 waves) per (b*h, 16-row query tile).
__global__ __launch_bounds__(128) void
LinearKernelAttention_82592221102306_kernel(const float* __restrict__ Q,
                                            const float* __restrict__ K,
                                            const float* __restrict__ V,
                                            float* __restrict__ Out,
                                            float* __restrict__ Attn) {
  extern __shared__ float scores[];                    // 16 x LDS_STRIDE raw masked scores
  __shared__ float part[4][16];                        // per-wave row sum-of-squares partials
  __shared__ float invn[16];                           // 1 / max(||row||, 1e-12)
  __shared__ __align__(16) float outpart[4][16][DIM];  // per-wave out partials

  const int tid   = threadIdx.x;
  const int wave  = tid >> 5;
  const int lane  = tid & 31;
  const int bh    = blockIdx.x >> 7;           // 0..15  (b*h)
  const int qtile = blockIdx.x & 127;          // 0..127 (16-row tiles)
  const int m0    = qtile * 16;

  const int lhalf = lane >> 4;                 // which half-wave
  const int l16   = lane & 15;
  const int kbase = lhalf * 2;                 // A/B frag: K offset per ISA layout

  const float* Qb = Q + (size_t)bh * SEQ * DIM;
  const float* Kb = K + (size_t)bh * SEQ * DIM;
  const float* Vb = V + (size_t)bh * SEQ * DIM;
  float* Ob = Out  + (size_t)bh * SEQ * DIM;
  float* Ab = Attn + (size_t)bh * SEQ * SEQ;

  // ---- A-fragments of Q (16 chunks of K=4 covering D=64), loaded once ----
  v2f aQ[16];
  {
    const float* qrow = Qb + (size_t)(m0 + l16) * DIM;
    #pragma unroll
    for (int kc = 0; kc < 16; ++kc) {
      aQ[kc].x = qrow[kc * 4 + kbase + 0];
      aQ[kc].y = qrow[kc * 4 + kbase + 1];
    }
  }

  float rowsq[8];
  #pragma unroll
  for (int r = 0; r < 8; ++r) rowsq[r] = 0.0f;

  // ---- Phase A: scores = Q*K^T + eps (masked), stash to LDS, sum of squares ----
  for (int j = wave; j <= qtile; j += 4) {
    v2f bK[16];
    const float* krow = Kb + (size_t)(j * 16 + l16) * DIM;
    #pragma unroll
    for (int kc = 0; kc < 16; ++kc) {
      bK[kc].x = krow[kc * 4 + kbase + 0];
      bK[kc].y = krow[kc * 4 + kbase + 1];
    }
    // two independent WMMA accumulation chains for ILP
    v8f acc0 = {0.f, 0.f, 0.f, 0.f, 0.f, 0.f, 0.f, 0.f};
    v8f acc1 = {0.f, 0.f, 0.f, 0.f, 0.f, 0.f, 0.f, 0.f};
    #pragma unroll
    for (int kc = 0; kc < 8; ++kc) {
      acc0 = __builtin_amdgcn_wmma_f32_16x16x4_f32(false, aQ[2 * kc + 0], false, bK[2 * kc + 0],
                                                   (short)0, acc0, false, false);
      acc1 = __builtin_amdgcn_wmma_f32_16x16x4_f32(false, aQ[2 * kc + 1], false, bK[2 * kc + 1],
                                                   (short)0, acc1, false, false);
    }

    #pragma unroll
    for (int vv = 0; vv < 8; ++vv) {
      const int r  = vv + lhalf * 8;           // row within tile (C/D layout)
      const int mg = m0 + r;
      const int ng = j * 16 + l16;
      float s = (ng <= mg) ? (acc0[vv] + acc1[vv] + 1e-5f) : 0.0f;
      rowsq[vv] += s * s;
      scores[r * LDS_STRIDE + j * 16 + l16] = s;
    }
  }

  // reduce sum-of-squares across the 16 lanes sharing each row
  #pragma unroll
  for (int vv = 0; vv < 8; ++vv) {
    float vsum = rowsq[vv];
    #pragma unroll
    for (int off = 8; off >= 1; off >>= 1)
      vsum += __shfl_xor(vsum, off, 16);
    if (l16 == 0) part[wave][vv + lhalf * 8] = vsum;
  }
  __syncthreads();
  if (tid < 16) {
    float sum = part[0][tid] + part[1][tid] + part[2][tid] + part[3][tid];
    float nrm = sqrtf(sum);
    invn[tid] = 1.0f / fmaxf(nrm, 1e-12f);
  }
  __syncthreads();

  // ---- zero-fill strictly-causal-masked attn tiles (j > qtile), NT streamed ----
  {
    const int zstart = (qtile + 1) * 16;
    const int zcols  = SEQ - zstart;           // multiple of 16
    if (zcols > 0) {
      const int c4count = zcols >> 2;
      const int r  = tid >> 3;                 // 16 rows, 8 threads each
      const int t8 = tid & 7;
      const v4f z = {0.f, 0.f, 0.f, 0.f};
      v4f* base = (v4f*)(Ab + (size_t)(m0 + r) * SEQ + zstart);
      for (int c4 = t8; c4 < c4count; c4 += 8)
        __builtin_nontemporal_store(z, base + c4);
    }
  }

  // ---- Phase B: stream normalized attn (NT, b128); out += attn * V via f32 WMMA ----
  v8f oacc[4];
  #pragma unroll
  for (int db = 0; db < 4; ++db)
    oacc[db] = (v8f){0.f, 0.f, 0.f, 0.f, 0.f, 0.f, 0.f, 0.f};

  for (int j = wave; j <= qtile; j += 4) {
    // 16x16 tile = 64 float4 slots; lanes cover 4 consecutive cols each (coalesced 64B/4 lanes)
    #pragma unroll
    for (int e = 0; e < 2; ++e) {
      const int idx = e * 32 + lane;           // 0..63
      const int r  = idx >> 2;
      const int c4 = (idx & 3) * 4;
      const v4f sv = *(const v4f*)&scores[r * LDS_STRIDE + j * 16 + c4];
      const float in = invn[r];
      const v4f val = sv * in;
      __builtin_nontemporal_store(val, (v4f*)(Ab + (size_t)(m0 + r) * SEQ + j * 16 + c4));
    }
    // attn tile (K=16) consumed as 4 chunks of K=4
    #pragma unroll
    for (int nc = 0; nc < 4; ++nc) {
      v2f a;
      const int m = l16;
      a.x = scores[m * LDS_STRIDE + j * 16 + nc * 4 + kbase + 0] * invn[m];
      a.y = scores[m * LDS_STRIDE + j * 16 + nc * 4 + kbase + 1] * invn[m];
      #pragma unroll
      for (int db = 0; db < 4; ++db) {
        v2f b;
        const float* vr = Vb + (size_t)(j * 16 + nc * 4 + kbase) * DIM + db * 16 + l16;
        b.x = vr[0];
        b.y = vr[DIM];
        oacc[db] = __builtin_amdgcn_wmma_f32_16x16x4_f32(false, a, false, b,
                                                         (short)0, oacc[db], false, false);
      }
    }
  }

  // ---- reduce the 4 per-wave out partials and store 16x64 out tile (b128) ----
  #pragma unroll
  for (int db = 0; db < 4; ++db)
    #pragma unroll
    for (int vv = 0; vv < 8; ++vv)
      outpart[wave][vv + lhalf * 8][db * 16 + l16] = oacc[db][vv];
  __syncthreads();
  for (int i = tid; i < 16 * 16; i += 128) {   // 256 float4 slots
    const int r  = i >> 4;
    const int c4 = (i & 15) * 4;
    const v4f a0 = *(const v4f*)&outpart[0][r][c4];
    const v4f a1 = *(const v4f*)&outpart[1][r][c4];
    const v4f a2 = *(const v4f*)&outpart[2][r][c4];
    const v4f a3 = *(const v4f*)&outpart[3][r][c4];
    const v4f s = a0 + a1 + a2 + a3;
    *(v4f*)(Ob + (size_t)(m0 + r) * DIM + c4) = s;
  }
}

extern "C" void kernel_launch(void* const* d_in, const int* in_sizes, int n_in,
                              void* d_out, int out_size, void* d_ws, size_t ws_size,
                              hipStream_t stream) {
  (void)in_sizes; (void)n_in; (void)out_size; (void)d_ws; (void)ws_size;
  const float* Q = (const float*)d_in[0];
  const float* K = (const float*)d_in[1];
  const float* V = (const float*)d_in[2];
  // d_in[3] is the causal mask; causality is applied analytically in-kernel.
  float* out  = (float*)d_out;
  float* attn = out + (size_t)2 * 8 * SEQ * DIM;   // attn follows out in d_out

  const dim3 grid(16 * 128);       // (b*h) x 128 query tiles
  const dim3 block(128);           // 4 wave32s
  const size_t shmem = (size_t)16 * LDS_STRIDE * sizeof(float);  // ~128 KB score stash
  LinearKernelAttention_82592221102306_kernel<<<grid, block, shmem, stream>>>(
      Q, K, V, out, attn);
}